// ProposalLayer_60885456388492
// MI455X (gfx1250) — compile-verified
//
#include <hip/hip_runtime.h>
#include <cstdint>

#define NB 8
#define NA 9
#define NH 128
#define NW 128
#define NHW (NH*NW)         // 16384 = 2^14
#define NP (NA*NHW)         // 147456 proposals per batch
#define PRE 6000
#define POST 300
#define SORTN 8192
#define FEATF 16.0f
#define NMS_T 0.7f
#define MIN_SZ 16.0f
#define NEGINF (-1e30f)

// float -> order-preserving uint32 key (ascending key == ascending float)
__device__ __forceinline__ uint32_t f2key(float f) {
    uint32_t u = __float_as_uint(f);
    return (u & 0x80000000u) ? ~u : (u | 0x80000000u);
}

// ---------------------------------------------------------------- K1: decode
// One thread per (b, h*W+w); loops over the 9 anchors. All global reads are
// lane-contiguous (coalesced); anchor data is loop-uniform (scalar loads).
__global__ void k_boxes(const float* __restrict__ scores,
                        const float* __restrict__ deltas,
                        const float* __restrict__ im_info,
                        const float* __restrict__ anchors,
                        float4* __restrict__ boxes,
                        uint32_t* __restrict__ keys)
{
    int t = blockIdx.x * blockDim.x + threadIdx.x;   // < NB*NHW
    int b  = t >> 14;
    int hw = t & (NHW - 1);
    int w  = hw & (NW - 1);
    int h  = hw >> 7;

    float sx = (float)w * FEATF;
    float sy = (float)h * FEATF;

    float imh = im_info[b*3+0];
    float imw = im_info[b*3+1];
    float scl = im_info[b*3+2];
    float ms  = MIN_SZ * scl;

    const float* dB = deltas + (size_t)b * (4*NA) * NHW + hw;
    const float* sB = scores + ((size_t)b * (2*NA) + NA) * NHW + hw;
    float4*   boxB  = boxes + (size_t)b * NP + (size_t)hw * NA;
    uint32_t* keyB  = keys  + (size_t)b * NP + (size_t)hw * NA;

#pragma unroll
    for (int a = 0; a < NA; ++a) {
        float ax1 = anchors[a*4+0] + sx;
        float ay1 = anchors[a*4+1] + sy;
        float ax2 = anchors[a*4+2] + sx;
        float ay2 = anchors[a*4+3] + sy;
        float aw  = ax2 - ax1 + 1.0f;
        float ah  = ay2 - ay1 + 1.0f;
        float acx = ax1 + 0.5f * aw;
        float acy = ay1 + 0.5f * ah;

        float dx = dB[(a*4 + 0) * NHW];
        float dy = dB[(a*4 + 1) * NHW];
        float dw = dB[(a*4 + 2) * NHW];
        float dh = dB[(a*4 + 3) * NHW];
        float s  = sB[a * NHW];

        float pcx = dx * aw + acx;
        float pcy = dy * ah + acy;
        float pw  = expf(dw) * aw;
        float ph  = expf(dh) * ah;
        float x1 = pcx - 0.5f * pw;
        float y1 = pcy - 0.5f * ph;
        float x2 = pcx + 0.5f * pw;
        float y2 = pcy + 0.5f * ph;

        x1 = fminf(fmaxf(x1, 0.0f), imw - 1.0f);
        y1 = fminf(fmaxf(y1, 0.0f), imh - 1.0f);
        x2 = fminf(fmaxf(x2, 0.0f), imw - 1.0f);
        y2 = fminf(fmaxf(y2, 0.0f), imh - 1.0f);

        bool valid = ((x2 - x1 + 1.0f) >= ms) && ((y2 - y1 + 1.0f) >= ms);
        if (!valid) s = NEGINF;

        boxB[a] = make_float4(x1, y1, x2, y2);   // reference order p = hw*9+a
        keyB[a] = f2key(s);
    }
}

// ------------------------------------------------- K2: exact radix top-k cut
__global__ __launch_bounds__(1024)
void k_radix(const uint32_t* __restrict__ keys,
             uint32_t* __restrict__ T, uint32_t* __restrict__ G,
             uint32_t* __restrict__ E)
{
    __shared__ uint32_t hist[256];
    __shared__ uint32_t s_v, s_run;
    int b = blockIdx.x, tid = threadIdx.x, bd = blockDim.x;
    const uint32_t* kb = keys + (size_t)b * NP;

    uint32_t needed = PRE, g = 0, pref = 0;
    for (int l = 3; l >= 0; --l) {
        for (int i = tid; i < 256; i += bd) hist[i] = 0;
        __syncthreads();
        for (int i = tid; i < NP; i += bd) {
            uint32_t k = kb[i];
            bool m = (l == 3) || ((k >> ((l + 1) * 8)) == pref);
            if (m) atomicAdd(&hist[(k >> (l * 8)) & 255u], 1u);
        }
        __syncthreads();
        if (tid == 0) {
            uint32_t run = 0, v = 0;
            for (int bin = 255; bin >= 0; --bin) {
                uint32_t c = hist[bin];
                if (run + c >= needed) { v = (uint32_t)bin; break; }
                run += c;
            }
            s_v = v; s_run = run;
        }
        __syncthreads();
        uint32_t v = s_v, run = s_run;
        g += run;            // keys strictly greater at this level
        needed -= run;
        pref = (pref << 8) | v;
        __syncthreads();
    }
    if (tid == 0) { T[b] = pref; G[b] = g; E[b] = needed; }
}

// ------------------------------------------------------------- K3: compact
__global__ void k_init(uint32_t* cnt) {
    int t = blockIdx.x * blockDim.x + threadIdx.x;
    if (t < NB * 2) cnt[t] = 0;
}

__global__ void k_compact(const uint32_t* __restrict__ keys,
                          const uint32_t* __restrict__ T,
                          const uint32_t* __restrict__ G,
                          const uint32_t* __restrict__ E,
                          uint32_t* __restrict__ cnt,
                          unsigned long long* __restrict__ sel)
{
    int t = blockIdx.x * blockDim.x + threadIdx.x;
    if (t >= NB * NP) return;
    int b = t / NP;
    uint32_t p = (uint32_t)(t - b * NP);
    uint32_t k = keys[t];
    uint32_t Tb = T[b];
    unsigned long long v = ((unsigned long long)k << 32) | (uint32_t)(~p);
    if (k > Tb) {
        uint32_t slot = atomicAdd(&cnt[b*2+0], 1u);
        sel[(size_t)b * PRE + slot] = v;
    } else if (k == Tb) {
        uint32_t ts = atomicAdd(&cnt[b*2+1], 1u);
        if (ts < E[b]) sel[(size_t)b * PRE + G[b] + ts] = v;
    }
}

// -------------------------------- K4: async-load to LDS + bitonic sort 8192
__global__ __launch_bounds__(1024)
void k_sort(const unsigned long long* __restrict__ sel,
            const float4* __restrict__ boxes,
            float4* __restrict__ sboxes)
{
    __shared__ unsigned long long a[SORTN];   // 64 KB LDS
    int b = blockIdx.x, tid = threadIdx.x;
    unsigned long long base = (unsigned long long)(uintptr_t)(sel + (size_t)b * PRE);

    // Async global->LDS staging of the 6000 selected (key,~idx) pairs.
    uint32_t lbase = (uint32_t)(uintptr_t)(&a[0]);
    for (int i = tid; i < PRE; i += 1024) {
        uint32_t ldsoff = lbase + (uint32_t)i * 8u;
        uint32_t voff   = (uint32_t)i * 8u;
        asm volatile("global_load_async_to_lds_b64 %0, %1, %2"
                     :: "v"(ldsoff), "v"(voff), "s"(base) : "memory");
    }
    for (int i = PRE + tid; i < SORTN; i += 1024) a[i] = 0ull;   // pad low
    asm volatile("s_wait_asynccnt 0" ::: "memory");
    __syncthreads();

    // Ascending bitonic sort of SORTN 64-bit composites.
    for (int k = 2; k <= SORTN; k <<= 1) {
        for (int j = k >> 1; j > 0; j >>= 1) {
            for (int i = tid; i < SORTN; i += 1024) {
                int ixj = i ^ j;
                if (ixj > i) {
                    unsigned long long x = a[i], y = a[ixj];
                    bool up = ((i & k) == 0);
                    if ((x > y) == up) { a[i] = y; a[ixj] = x; }
                }
            }
            __syncthreads();
        }
    }

    // Emit top PRE in descending score order; gather boxes by index.
    for (int r = tid; r < PRE; r += 1024) {
        unsigned long long v = a[SORTN - 1 - r];
        uint32_t idx = ~(uint32_t)(v & 0xffffffffull);
        sboxes[(size_t)b * PRE + r] = boxes[(size_t)b * NP + idx];
    }
}

// ----------------------------------------- K5: wave32 greedy NMS, LDS cache
__global__ void k_nms(const float4* __restrict__ sboxes,
                      uint32_t* __restrict__ keep)
{
    __shared__ float4 kept[PRE];              // 96 KB LDS kept-box cache
    int b = blockIdx.x, lane = threadIdx.x;
    const float4* sb = sboxes + (size_t)b * PRE;
    uint32_t* kp = keep + (size_t)b * PRE;

    int nk = 0;   // uniform across the wave
    for (int r = 0; r < PRE; ++r) {
        float4 bx = sb[r];
        float barea = (bx.z - bx.x + 1.0f) * (bx.w - bx.y + 1.0f);
        bool sup = false;
        for (int base = 0; base < nk; base += 32) {
            int kk = base + lane;
            bool pred = false;
            if (kk < nk) {
                float4 kb = kept[kk];
                float karea = (kb.z - kb.x + 1.0f) * (kb.w - kb.y + 1.0f);
                float xx1 = fmaxf(bx.x, kb.x);
                float yy1 = fmaxf(bx.y, kb.y);
                float xx2 = fminf(bx.z, kb.z);
                float yy2 = fminf(bx.w, kb.w);
                float iw = fmaxf(0.0f, xx2 - xx1 + 1.0f);
                float ih = fmaxf(0.0f, yy2 - yy1 + 1.0f);
                float inter = iw * ih;
                float iou = inter / (barea + karea - inter);
                pred = iou > NMS_T;
            }
            if (__ballot(pred) != 0ull) { sup = true; break; }
        }
        if (!sup) {
            if (lane == 0) kept[nk] = bx;
            nk++;
        }
        if (lane == 0) kp[r] = sup ? 0u : 1u;
        __syncthreads();   // single-wave WG: cheap; makes LDS append visible
    }
}

// ---------------------------------------------------------- K6: finalize 300
__global__ void k_final(const float4* __restrict__ sboxes,
                        const uint32_t* __restrict__ keep,
                        float* __restrict__ out)
{
    __shared__ int selidx[POST];
    __shared__ int s_cnt;
    int b = blockIdx.x, tid = threadIdx.x;
    if (tid == 0) {
        int c = 0;
        for (int r = 0; r < PRE && c < POST; ++r)
            if (keep[(size_t)b * PRE + r]) selidx[c++] = r;
        s_cnt = c;
    }
    __syncthreads();
    int c = s_cnt;
    for (int j = tid; j < POST; j += blockDim.x) {
        float* o = out + ((size_t)b * POST + j) * 5;
        if (j < c) {
            float4 bx = sboxes[(size_t)b * PRE + selidx[j]];
            o[0] = (float)b; o[1] = bx.x; o[2] = bx.y; o[3] = bx.z; o[4] = bx.w;
        } else {
            o[0] = (float)b; o[1] = 0.0f; o[2] = 0.0f; o[3] = 0.0f; o[4] = 0.0f;
        }
    }
}

// ------------------------------------------------------------------- launch
extern "C" void kernel_launch(void* const* d_in, const int* in_sizes, int n_in,
                              void* d_out, int out_size, void* d_ws, size_t ws_size,
                              hipStream_t stream)
{
    const float* scores  = (const float*)d_in[0];   // (8, 18, 128, 128)
    const float* deltas  = (const float*)d_in[1];   // (8, 36, 128, 128)
    const float* im_info = (const float*)d_in[2];   // (8, 3)
    const float* anchors = (const float*)d_in[3];   // (9, 4)
    float* out = (float*)d_out;                     // (8, 300, 5)

    char* w = (char*)d_ws;
    auto alloc = [&](size_t bytes) -> void* {
        void* p = (void*)w;
        w += (bytes + 255) & ~(size_t)255;
        return p;
    };
    float4*             boxes  = (float4*)             alloc(sizeof(float4)   * NB * NP);
    uint32_t*           keys   = (uint32_t*)           alloc(sizeof(uint32_t) * NB * NP);
    uint32_t*           Tc     = (uint32_t*)           alloc(sizeof(uint32_t) * NB);
    uint32_t*           Gc     = (uint32_t*)           alloc(sizeof(uint32_t) * NB);
    uint32_t*           Ec     = (uint32_t*)           alloc(sizeof(uint32_t) * NB);
    uint32_t*           cnt    = (uint32_t*)           alloc(sizeof(uint32_t) * NB * 2);
    unsigned long long* sel    = (unsigned long long*) alloc(sizeof(unsigned long long) * NB * PRE);
    float4*             sboxes = (float4*)             alloc(sizeof(float4)   * NB * PRE);
    uint32_t*           keep   = (uint32_t*)           alloc(sizeof(uint32_t) * NB * PRE);

    k_init<<<1, 64, 0, stream>>>(cnt);
    k_boxes<<<(NB * NHW + 255) / 256, 256, 0, stream>>>(scores, deltas, im_info, anchors, boxes, keys);
    k_radix<<<NB, 1024, 0, stream>>>(keys, Tc, Gc, Ec);
    k_compact<<<(NB * NP + 255) / 256, 256, 0, stream>>>(keys, Tc, Gc, Ec, cnt, sel);
    k_sort<<<NB, 1024, 0, stream>>>(sel, boxes, sboxes);
    k_nms<<<NB, 32, 0, stream>>>(sboxes, keep);
    k_final<<<NB, 256, 0, stream>>>(sboxes, keep, out);
}